// WiKG_9869834847030
// MI455X (gfx1250) — compile-verified
//
#include <hip/hip_runtime.h>
#include <math.h>

typedef __attribute__((ext_vector_type(2))) float v2f;
typedef __attribute__((ext_vector_type(8))) float v8f;

#define NROWS 8192
#define DIN   384
#define DH    512
#define TOPK  6

__device__ __forceinline__ float leakyf(float v) { return v > 0.f ? v : 0.01f * v; }

__device__ __forceinline__ v8f wmma_f32(v2f a, v2f b, v8f c) {
    // D = A(16x4 f32) * B(4x16 f32) + C(16x16 f32)
    return __builtin_amdgcn_wmma_f32_16x16x4_f32(false, a, false, b, (short)0, c, false, false);
}

__device__ __forceinline__ float wred(float v) {
    #pragma unroll
    for (int off = 16; off > 0; off >>= 1) v += __shfl_xor(v, off, 32);
    return v;
}

// keep list sorted desc by (value, then index ascending) == jax.lax.top_k order
__device__ __forceinline__ void topk_insert(float* tv, int* ti, float val, int gi) {
    if ((val > tv[TOPK - 1]) || (val == tv[TOPK - 1] && gi < ti[TOPK - 1])) {
        tv[TOPK - 1] = val; ti[TOPK - 1] = gi;
        #pragma unroll
        for (int j = TOPK - 1; j > 0; --j) {
            bool sw = (tv[j] > tv[j - 1]) || (tv[j] == tv[j - 1] && ti[j] < ti[j - 1]);
            if (sw) {
                float fv = tv[j]; tv[j] = tv[j - 1]; tv[j - 1] = fv;
                int ii = ti[j]; ti[j] = ti[j - 1]; ti[j - 1] = ii;
            }
        }
    }
}

// ---------------------------------------------------------------------------
// Generic C = act(A[M x K] @ W[K x N] + bias) via V_WMMA_F32_16X16X4_F32.
// Block = 256 threads (8 waves): 16 rows x 128 cols per block. A tile in LDS.
// ---------------------------------------------------------------------------
__global__ __launch_bounds__(256) void gemm_wmma_kernel(
    const float* __restrict__ A, const float* __restrict__ W,
    const float* __restrict__ bias, float* __restrict__ C,
    int K, int N, int act, int accum)
{
    __shared__ float As[16 * 520];
    const int tid = threadIdx.x;
    const int mBase = blockIdx.x * 16;
    for (int i = tid; i < 16 * K; i += 256) {
        int r = i / K, k = i - r * K;
        As[r * 520 + k] = A[(mBase + r) * K + k];
    }
    __syncthreads();

    const int wave = tid >> 5, lane = tid & 31;
    const int l15 = lane & 15, kh = lane >> 4;
    const int nBase = blockIdx.y * 128 + wave * 16;
    const int col = nBase + l15;

    v8f c = {};
    const float* Wp = W + (2 * kh) * N + col;     // advance 4*N per k-step
    const float* Ap = As + l15 * 520 + 2 * kh;
    #pragma unroll 4
    for (int k0 = 0; k0 < K; k0 += 4) {
        v2f a = *(const v2f*)(Ap + k0);
        v2f b; b.x = Wp[0]; b.y = Wp[N];
        Wp += 4 * N;
        c = wmma_f32(a, b, c);
    }

    const float bv = bias[col];
    float* Cp = C + (mBase + 8 * kh) * N + col;
    #pragma unroll
    for (int v = 0; v < 8; ++v) {
        float val = c[v] + bv;
        if (act) val = leakyf(val);
        if (accum) val += Cp[v * N];
        Cp[v * N] = val;
    }
}

// ---------------------------------------------------------------------------
// Attention: logits = (e_h * 512^-0.5) @ e_t^T, streaming; keep top-6 per row.
// Block = 16 rows, 8 waves sweep the 512 column-tiles round-robin.
// ---------------------------------------------------------------------------
__global__ __launch_bounds__(256) void attn_topk_kernel(
    const float* __restrict__ eh, const float* __restrict__ et,
    float* __restrict__ tkw, int* __restrict__ tki)
{
    __shared__ float As[16 * 520];
    __shared__ float Cs[8 * 16 * 17];
    __shared__ float mvS[8 * 16 * TOPK];
    __shared__ int   miS[8 * 16 * TOPK];

    const int tid = threadIdx.x;
    const int mBase = blockIdx.x * 16;
    const float scale = 0.044194173824159216f;  // 512^-0.5

    for (int i = tid; i < 16 * DH; i += 256) {
        int r = i >> 9, k = i & (DH - 1);
        As[r * 520 + k] = eh[(mBase + r) * DH + k] * scale;
    }
    __syncthreads();

    const int wave = tid >> 5, lane = tid & 31;
    const int l15 = lane & 15, kh = lane >> 4;

    float tv[TOPK]; int ti[TOPK];
    #pragma unroll
    for (int j = 0; j < TOPK; ++j) { tv[j] = -INFINITY; ti[j] = 0x7fffffff; }

    float* Cw = Cs + wave * (16 * 17);
    const float* Ap = As + l15 * 520 + 2 * kh;

    for (int nt = wave; nt < NROWS / 16; nt += 8) {
        const int nBase = nt * 16;
        v8f c = {};
        const float* Bp = et + (nBase + l15) * DH + 2 * kh;  // B[k][n] = e_t[n][k]
        #pragma unroll 4
        for (int k0 = 0; k0 < DH; k0 += 4) {
            v2f a = *(const v2f*)(Ap + k0);
            v2f b = *(const v2f*)(Bp + k0);
            c = wmma_f32(a, b, c);
        }
        #pragma unroll
        for (int v = 0; v < 8; ++v) Cw[(v + 8 * kh) * 17 + l15] = c[v];
        // DS store->load same wave: hardware DScnt ordering handles the hazard
        if (lane < 16) {
            #pragma unroll
            for (int colq = 0; colq < 16; ++colq) {
                topk_insert(tv, ti, Cw[lane * 17 + colq], nBase + colq);
            }
        }
    }

    if (lane < 16) {
        #pragma unroll
        for (int j = 0; j < TOPK; ++j) {
            mvS[(wave * 16 + lane) * TOPK + j] = tv[j];
            miS[(wave * 16 + lane) * TOPK + j] = ti[j];
        }
    }
    __syncthreads();

    if (wave == 0 && lane < 16) {
        const int row = lane;
        float fv[TOPK]; int fi[TOPK];
        #pragma unroll
        for (int j = 0; j < TOPK; ++j) { fv[j] = -INFINITY; fi[j] = 0x7fffffff; }
        for (int w = 0; w < 8; ++w) {
            #pragma unroll
            for (int j = 0; j < TOPK; ++j) {
                topk_insert(fv, fi, mvS[(w * 16 + row) * TOPK + j],
                                     miS[(w * 16 + row) * TOPK + j]);
            }
        }
        #pragma unroll
        for (int j = 0; j < TOPK; ++j) {
            tkw[(mBase + row) * TOPK + j] = fv[j];
            tki[(mBase + row) * TOPK + j] = fi[j];
        }
    }
}

// ---------------------------------------------------------------------------
// Column mean of x (8192 x 512): partial per 128-row chunk, then finish.
// ---------------------------------------------------------------------------
__global__ __launch_bounds__(256) void colsum_partial_kernel(
    const float* __restrict__ x, float* __restrict__ partial)
{
    const int b = blockIdx.x;  // 64 blocks x 128 rows
    for (int c = threadIdx.x; c < DH; c += 256) {
        float s = 0.f;
        const int r0 = b * 128;
        for (int r = 0; r < 128; ++r) s += x[(r0 + r) * DH + c];
        partial[b * DH + c] = s;
    }
}

__global__ __launch_bounds__(256) void colmean_finish_kernel(
    const float* __restrict__ partial, float* __restrict__ cmean)
{
    const int c = blockIdx.x * 256 + threadIdx.x;  // 512 threads
    float s = 0.f;
    for (int b = 0; b < 64; ++b) s += partial[b * DH + c];
    cmean[c] = s * (1.f / NROWS);
}

__global__ __launch_bounds__(256) void apply_mean_kernel(
    float* __restrict__ x, const float* __restrict__ cmean)
{
    const int i = blockIdx.x * 256 + threadIdx.x;
    x[i] = (x[i] + cmean[i & (DH - 1)]) * 0.5f;
}

// ---------------------------------------------------------------------------
// Per-row combine: topk softmax, gate, ka softmax, e_Nh; one wave per row.
// Emits sumIn = e_h + e_Nh and mulIn = e_h * e_Nh for the Wl1/Wl2 GEMMs.
// ---------------------------------------------------------------------------
__global__ __launch_bounds__(256) void combine_kernel(
    const float* __restrict__ eh, const float* __restrict__ et,
    const float* __restrict__ tkw, const int* __restrict__ tki,
    float* __restrict__ sumIn, float* __restrict__ mulIn)
{
    const int wave = threadIdx.x >> 5, lane = threadIdx.x & 31;
    const int row = blockIdx.x * 8 + wave;

    float w[TOPK]; int ix[TOPK];
    #pragma unroll
    for (int t = 0; t < TOPK; ++t) { w[t] = tkw[row * TOPK + t]; ix[t] = tki[row * TOPK + t]; }

    float mx = w[0];
    #pragma unroll
    for (int t = 1; t < TOPK; ++t) mx = fmaxf(mx, w[t]);
    float p[TOPK], s = 0.f;
    #pragma unroll
    for (int t = 0; t < TOPK; ++t) { p[t] = expf(w[t] - mx); s += p[t]; }
    const float invs = 1.f / s;
    #pragma unroll
    for (int t = 0; t < TOPK; ++t) p[t] *= invs;

    float ehreg[16];
    #pragma unroll
    for (int j = 0; j < 16; ++j) ehreg[j] = eh[row * DH + lane + 32 * j];

    float kaw[TOPK];
    #pragma unroll
    for (int k = 0; k < TOPK; ++k) {
        const float* Nb = et + (long)ix[k] * DH;
        const float c1 = 2.f - p[k], c2 = p[k];   // tanh((2-p)*eh + p*Nb)
        float pn = 0.f, pg = 0.f;
        #pragma unroll
        for (int j = 0; j < 16; ++j) {
            float nb = Nb[lane + 32 * j];
            pn += nb;
            pg += tanhf(c1 * ehreg[j] + c2 * nb);
        }
        pn = wred(pn); pg = wred(pg);
        kaw[k] = pn * pg;
    }

    float km = kaw[0];
    #pragma unroll
    for (int t = 1; t < TOPK; ++t) km = fmaxf(km, kaw[t]);
    float kp[TOPK], ks = 0.f;
    #pragma unroll
    for (int t = 0; t < TOPK; ++t) { kp[t] = expf(kaw[t] - km); ks += kp[t]; }
    const float invk = 1.f / ks;
    #pragma unroll
    for (int t = 0; t < TOPK; ++t) kp[t] *= invk;

    #pragma unroll
    for (int j = 0; j < 16; ++j) {
        const int d = lane + 32 * j;
        float eN = 0.f;
        #pragma unroll
        for (int k = 0; k < TOPK; ++k) eN += kp[k] * et[(long)ix[k] * DH + d];
        const float ehd = ehreg[j];
        sumIn[row * DH + d] = ehd + eN;
        mulIn[row * DH + d] = ehd * eN;
    }
}

// scores = t1[8192 x 256] @ Wa2[256] + ba2 ; one wave per row
__global__ __launch_bounds__(256) void score_kernel(
    const float* __restrict__ t1, const float* __restrict__ Wa2,
    const float* __restrict__ ba2, float* __restrict__ scores)
{
    const int wave = threadIdx.x >> 5, lane = threadIdx.x & 31;
    const int row = blockIdx.x * 8 + wave;
    float s = 0.f;
    #pragma unroll
    for (int j = 0; j < 8; ++j) s += t1[row * 256 + lane + 32 * j] * Wa2[lane + 32 * j];
    s = wred(s);
    if (lane == 0) scores[row] = s + ba2[0];
}

// softmax over the 8192 scores, in place (single block, deterministic order)
__global__ __launch_bounds__(256) void softmax_norm_kernel(float* __restrict__ scores)
{
    __shared__ float red[256];
    const int tid = threadIdx.x;
    float m = -INFINITY;
    for (int r = tid; r < NROWS; r += 256) m = fmaxf(m, scores[r]);
    red[tid] = m; __syncthreads();
    for (int o = 128; o > 0; o >>= 1) {
        if (tid < o) red[tid] = fmaxf(red[tid], red[tid + o]);
        __syncthreads();
    }
    const float mx = red[0];
    __syncthreads();
    float s = 0.f;
    for (int r = tid; r < NROWS; r += 256) s += expf(scores[r] - mx);
    red[tid] = s; __syncthreads();
    for (int o = 128; o > 0; o >>= 1) {
        if (tid < o) red[tid] += red[tid + o];
        __syncthreads();
    }
    const float inv = 1.f / red[0];
    for (int r = tid; r < NROWS; r += 256) scores[r] = expf(scores[r] - mx) * inv;
}

// pooled[d] = sum_r alpha[r] * h[r][d]   (512 threads over 2 blocks)
__global__ __launch_bounds__(256) void pooled_kernel(
    const float* __restrict__ h, const float* __restrict__ alpha,
    float* __restrict__ pooled)
{
    const int d = blockIdx.x * 256 + threadIdx.x;
    float acc = 0.f;
    for (int r = 0; r < NROWS; ++r) acc += alpha[r] * h[r * DH + d];
    pooled[d] = acc;
}

// layernorm + fc + softmax + argmax -> out[5] = {l0, l1, p0, p1, yhat}
__global__ __launch_bounds__(256) void final_kernel(
    const float* __restrict__ pooled, const float* __restrict__ gamma,
    const float* __restrict__ beta, const float* __restrict__ Wfc,
    const float* __restrict__ bfc, float* __restrict__ out)
{
    __shared__ float red[256];
    const int tid = threadIdx.x;
    const float v0 = pooled[tid], v1 = pooled[tid + 256];

    red[tid] = v0 + v1; __syncthreads();
    for (int o = 128; o > 0; o >>= 1) { if (tid < o) red[tid] += red[tid + o]; __syncthreads(); }
    const float mu = red[0] * (1.f / DH);
    __syncthreads();

    const float d0 = v0 - mu, d1 = v1 - mu;
    red[tid] = d0 * d0 + d1 * d1; __syncthreads();
    for (int o = 128; o > 0; o >>= 1) { if (tid < o) red[tid] += red[tid + o]; __syncthreads(); }
    const float var = red[0] * (1.f / DH);
    const float rstd = rsqrtf(var + 1e-5f);
    __syncthreads();

    const float p0 = d0 * rstd * gamma[tid] + beta[tid];
    const float p1 = d1 * rstd * gamma[tid + 256] + beta[tid + 256];

    red[tid] = p0 * Wfc[tid * 2 + 0] + p1 * Wfc[(tid + 256) * 2 + 0]; __syncthreads();
    for (int o = 128; o > 0; o >>= 1) { if (tid < o) red[tid] += red[tid + o]; __syncthreads(); }
    const float L0 = red[0] + bfc[0];
    __syncthreads();
    red[tid] = p0 * Wfc[tid * 2 + 1] + p1 * Wfc[(tid + 256) * 2 + 1]; __syncthreads();
    for (int o = 128; o > 0; o >>= 1) { if (tid < o) red[tid] += red[tid + o]; __syncthreads(); }
    const float L1 = red[0] + bfc[1];

    if (tid == 0) {
        const float mxl = fmaxf(L0, L1);
        const float e0 = expf(L0 - mxl), e1 = expf(L1 - mxl);
        const float si = 1.f / (e0 + e1);
        out[0] = L0; out[1] = L1;
        out[2] = e0 * si; out[3] = e1 * si;
        out[4] = (L1 > L0) ? 1.f : 0.f;   // argmax, first max on tie
    }
}

// ---------------------------------------------------------------------------
extern "C" void kernel_launch(void* const* d_in, const int* in_sizes, int n_in,
                              void* d_out, int out_size, void* d_ws, size_t ws_size,
                              hipStream_t stream)
{
    (void)in_sizes; (void)n_in; (void)out_size; (void)ws_size;
    const float* data = (const float*)d_in[0];
    // d_in[1] (CT_data) is dead in the reference graph -> skipped
    const float* W1  = (const float*)d_in[2];
    const float* b1  = (const float*)d_in[3];
    const float* Wh  = (const float*)d_in[4];
    const float* bh  = (const float*)d_in[5];
    const float* Wt  = (const float*)d_in[6];
    const float* bt  = (const float*)d_in[7];
    const float* Wl1 = (const float*)d_in[8];
    const float* bl1 = (const float*)d_in[9];
    const float* Wl2 = (const float*)d_in[10];
    const float* bl2 = (const float*)d_in[11];
    const float* Wa1 = (const float*)d_in[12];
    const float* ba1 = (const float*)d_in[13];
    const float* Wa2 = (const float*)d_in[14];
    const float* ba2 = (const float*)d_in[15];
    const float* gam = (const float*)d_in[16];
    const float* bet = (const float*)d_in[17];
    const float* Wfc = (const float*)d_in[18];
    const float* bfc = (const float*)d_in[19];

    float* ws = (float*)d_ws;
    float* x      = ws;                 // 8192*512 ; reused as sumIn after e_h/e_t
    float* ehp    = ws + 4194304;       // 8192*512
    float* etp    = ws + 8388608;       // 8192*512
    float* mulIn  = ws + 12582912;      // 8192*512
    float* hbuf   = ws + 16777216;      // 8192*512
    float* t1     = ws + 20971520;      // 8192*256
    float* part   = ws + 23068672;      // 64*512
    float* cmean  = ws + 23101440;      // 512
    float* tkw    = ws + 23101952;      // 8192*6
    int*   tki    = (int*)(ws + 23151104); // 8192*6
    float* scores = ws + 23200256;      // 8192
    float* pooled = ws + 23208448;      // 512

    // x = leaky(data @ W1 + b1)
    gemm_wmma_kernel<<<dim3(NROWS / 16, DH / 128), 256, 0, stream>>>(data, W1, b1, x, DIN, DH, 1, 0);
    // x = (x + mean_rows(x)) * 0.5
    colsum_partial_kernel<<<64, 256, 0, stream>>>(x, part);
    colmean_finish_kernel<<<2, 256, 0, stream>>>(part, cmean);
    apply_mean_kernel<<<NROWS * DH / 256, 256, 0, stream>>>(x, cmean);
    // e_h, e_t
    gemm_wmma_kernel<<<dim3(NROWS / 16, DH / 128), 256, 0, stream>>>(x, Wh, bh, ehp, DH, DH, 0, 0);
    gemm_wmma_kernel<<<dim3(NROWS / 16, DH / 128), 256, 0, stream>>>(x, Wt, bt, etp, DH, DH, 0, 0);
    // attention logits + top-6 (streamed, never materialized)
    attn_topk_kernel<<<NROWS / 16, 256, 0, stream>>>(ehp, etp, tkw, tki);
    // gate / ka softmax / e_Nh  ->  sumIn (reuses x), mulIn
    combine_kernel<<<NROWS / 8, 256, 0, stream>>>(ehp, etp, tkw, tki, x, mulIn);
    // h = leaky(sumIn@Wl1+bl1) + leaky(mulIn@Wl2+bl2)
    gemm_wmma_kernel<<<dim3(NROWS / 16, DH / 128), 256, 0, stream>>>(x, Wl1, bl1, hbuf, DH, DH, 1, 0);
    gemm_wmma_kernel<<<dim3(NROWS / 16, DH / 128), 256, 0, stream>>>(mulIn, Wl2, bl2, hbuf, DH, DH, 1, 1);
    // attention pooling head
    gemm_wmma_kernel<<<dim3(NROWS / 16, 256 / 128), 256, 0, stream>>>(hbuf, Wa1, ba1, t1, DH, 256, 1, 0);
    score_kernel<<<NROWS / 8, 256, 0, stream>>>(t1, Wa2, ba2, scores);
    softmax_norm_kernel<<<1, 256, 0, stream>>>(scores);
    pooled_kernel<<<2, 256, 0, stream>>>(hbuf, scores, pooled);
    final_kernel<<<1, 256, 0, stream>>>(pooled, gam, bet, Wfc, bfc, (float*)d_out);
}